// HeuristicGNN_42838003810334
// MI455X (gfx1250) — compile-verified
//
#include <hip/hip_runtime.h>
#include <hip/hip_bf16.h>
#include <math.h>

#define N_NODES 100000
#define N_EDGES 1600000
#define HID 64
#define DEPTH 3

typedef __attribute__((ext_vector_type(2))) float v2f;
typedef __attribute__((ext_vector_type(8))) float v8f;

// ---------------------------------------------------------------------------
// Node encoder: x[n][h] = b_node[h] + sum_k nf[n][k] * W_node[h][k]   (K=3)
// ---------------------------------------------------------------------------
__global__ void encode_nodes_kernel(const float* __restrict__ nf,
                                    const float* __restrict__ Wn,
                                    const float* __restrict__ bn,
                                    float* __restrict__ x) {
  int t = blockIdx.x * blockDim.x + threadIdx.x;
  if (t >= N_NODES * HID) return;
  int n = t >> 6, h = t & 63;
  float f0 = nf[n * 3 + 0], f1 = nf[n * 3 + 1], f2 = nf[n * 3 + 2];
  x[t] = bn[h] + Wn[h * 3 + 0] * f0 + Wn[h * 3 + 1] * f1 + Wn[h * 3 + 2] * f2;
}

__global__ void zero_kernel(float* __restrict__ p, int n) {
  int t = blockIdx.x * blockDim.x + threadIdx.x;
  if (t < n) p[t] = 0.0f;
}

// ---------------------------------------------------------------------------
// Scatter: one thread per (edge, feature).  A wave covers 32 consecutive
// features of one edge -> coalesced x[src] read + coalesced atomic segment;
// src/dst/attr loads are wave-uniform (broadcast).  e recomputed on the fly
// (3 FMAs/lane) instead of streaming a 409 MB materialized e array.
// agg (25.6 MB) is L2-resident -> atomics stay on-chip.
// ---------------------------------------------------------------------------
__global__ void scatter_msg_kernel(const float* __restrict__ x,
                                   const float* __restrict__ attr,
                                   const int* __restrict__ ei,
                                   const float* __restrict__ We,
                                   const float* __restrict__ be,
                                   float* __restrict__ agg) {
  long t = (long)blockIdx.x * blockDim.x + threadIdx.x;
  if (t >= (long)N_EDGES * HID) return;
  int e = (int)(t >> 6);
  int f = (int)(t & 63);
  int s = ei[e];
  int d = ei[N_EDGES + e];
  float a0 = attr[e * 3 + 0], a1 = attr[e * 3 + 1], a2 = attr[e * 3 + 2];
  float ev = be[f] + We[f * 3 + 0] * a0 + We[f * 3 + 1] * a1 + We[f * 3 + 2] * a2;
  float msg = x[(long)s * HID + f] + ev;
  atomicAdd(&agg[(long)d * HID + f], msg);
}

// ---------------------------------------------------------------------------
// GRU cell: one wave (32 threads) computes 16 nodes.
//   gi = agg @ W_ih^T + b_ih   [16 x 192]
//   gh = x   @ W_hh^T + b_hh   [16 x 192]
// via V_WMMA_F32_16X16X4_F32 (fp32 WMMA), then fused gating.
// ---------------------------------------------------------------------------
__global__ __launch_bounds__(32) void gru_kernel(const float* __restrict__ agg,
                                                 const float* __restrict__ x,
                                                 const float* __restrict__ Wih,
                                                 const float* __restrict__ bih,
                                                 const float* __restrict__ Whh,
                                                 const float* __restrict__ bhh,
                                                 float* __restrict__ xnew) {
  __shared__ float Ai[16][HID];   // agg tile
  __shared__ float Ah[16][HID];   // x (hidden) tile
  __shared__ float gi[16][192];
  __shared__ float gh[16][192];

  const int nb   = blockIdx.x * 16;
  const int lane = threadIdx.x;
  const int row  = lane & 15;     // M row / N col of this lane
  const int hi   = lane >> 4;     // 0: K=0,1 half, 1: K=2,3 half

  // stage A tiles into LDS (16*64 elems each, 32 lanes -> 32 iters)
  for (int it = 0; it < 32; ++it) {
    int idx = it * 32 + lane;
    int m = idx >> 6, f = idx & 63;
    Ai[m][f] = agg[(long)(nb + m) * HID + f];
    Ah[m][f] = x[(long)(nb + m) * HID + f];
  }
  __syncthreads();

  for (int which = 0; which < 2; ++which) {
    const float* W = which ? Whh : Wih;
    const float* B = which ? bhh : bih;
    for (int ob = 0; ob < 12; ++ob) {          // 12 x 16 = 192 output dims
      float bias = B[ob * 16 + row];
      v8f c = {bias, bias, bias, bias, bias, bias, bias, bias};
      for (int t = 0; t < 16; ++t) {           // K = 64 in steps of 4
        int k = t * 4 + hi * 2;
        v2f a, b;
        if (which) { a.x = Ah[row][k]; a.y = Ah[row][k + 1]; }
        else       { a.x = Ai[row][k]; a.y = Ai[row][k + 1]; }
        b.x = W[(ob * 16 + row) * HID + k];
        b.y = W[(ob * 16 + row) * HID + k + 1];
        c = __builtin_amdgcn_wmma_f32_16x16x4_f32(false, a, false, b,
                                                  (short)0, c, false, false);
      }
#pragma unroll
      for (int v = 0; v < 8; ++v) {
        int m = v + 8 * hi;
        if (which) gh[m][ob * 16 + row] = c[v];
        else       gi[m][ob * 16 + row] = c[v];
      }
    }
  }
  __syncthreads();

  // fused gates: r = sig(i_r+h_r), z = sig(i_z+h_z), n = tanh(i_n + r*h_n)
  for (int it = 0; it < 32; ++it) {
    int idx = it * 32 + lane;
    int m = idx >> 6, f = idx & 63;
    float ir = gi[m][f], iz = gi[m][64 + f], in_ = gi[m][128 + f];
    float hr = gh[m][f], hz = gh[m][64 + f], hn  = gh[m][128 + f];
    float r = 1.0f / (1.0f + __expf(-(ir + hr)));
    float z = 1.0f / (1.0f + __expf(-(iz + hz)));
    float n = tanhf(in_ + r * hn);
    float h = Ah[m][f];
    xnew[(long)(nb + m) * HID + f] = (1.0f - z) * n + z * h;
  }
}

// ---------------------------------------------------------------------------
// Decoder: one wave per 16 edges.
//   Acat = [x[src], x[dst], e]  (e recomputed from attr)      [16 x 192]
//   H    = relu(Acat @ W_dec1^T + b_dec1)                     [16 x 64]
//   out  = softplus(H @ W_dec2^T + b_dec2)                    [16]
// ---------------------------------------------------------------------------
__global__ __launch_bounds__(32) void decoder_kernel(const float* __restrict__ x,
                                                     const float* __restrict__ attr,
                                                     const int* __restrict__ ei,
                                                     const float* __restrict__ We,
                                                     const float* __restrict__ be,
                                                     const float* __restrict__ Wd1,
                                                     const float* __restrict__ bd1,
                                                     const float* __restrict__ Wd2,
                                                     const float* __restrict__ bd2,
                                                     float* __restrict__ out) {
  __shared__ float A[16][192];
  __shared__ float H[16][HID];

  const int eb   = blockIdx.x * 16;
  const int lane = threadIdx.x;
  const int row  = lane & 15;
  const int hi   = lane >> 4;

  // stage concat features (x gathers hit L2; e recomputed, K=3)
  for (int i = 0; i < 16; ++i) {
    int e = eb + i;
    int s = ei[e];
    int d = ei[N_EDGES + e];
    float a0 = attr[e * 3 + 0], a1 = attr[e * 3 + 1], a2 = attr[e * 3 + 2];
#pragma unroll
    for (int st = 0; st < 6; ++st) {
      int f = st * 32 + lane;
      float v;
      if (f < 64) {
        v = x[(long)s * HID + f];
      } else if (f < 128) {
        v = x[(long)d * HID + (f - 64)];
      } else {
        int h = f - 128;
        v = be[h] + We[h * 3 + 0] * a0 + We[h * 3 + 1] * a1 + We[h * 3 + 2] * a2;
      }
      A[i][f] = v;
    }
  }
  __syncthreads();

  for (int ob = 0; ob < 4; ++ob) {             // 4 x 16 = 64 output dims
    float bias = bd1[ob * 16 + row];
    v8f c = {bias, bias, bias, bias, bias, bias, bias, bias};
    for (int t = 0; t < 48; ++t) {             // K = 192 in steps of 4
      int k = t * 4 + hi * 2;
      v2f a, b;
      a.x = A[row][k];
      a.y = A[row][k + 1];
      b.x = Wd1[(ob * 16 + row) * 192 + k];
      b.y = Wd1[(ob * 16 + row) * 192 + k + 1];
      c = __builtin_amdgcn_wmma_f32_16x16x4_f32(false, a, false, b,
                                                (short)0, c, false, false);
    }
#pragma unroll
    for (int v = 0; v < 8; ++v)
      H[v + 8 * hi][ob * 16 + row] = fmaxf(c[v], 0.0f);
  }
  __syncthreads();

  // per-edge reduction against W_dec2: lanes 0-15 handle edges, two K-halves
  int i = row;
  float sum = 0.0f;
#pragma unroll
  for (int f = 0; f < 32; ++f)
    sum += H[i][hi * 32 + f] * Wd2[hi * 32 + f];
  sum += __shfl_xor(sum, 16, 32);
  if (lane < 16) {
    float t = sum + bd2[0];
    out[eb + i] = (t > 20.0f) ? t : log1pf(__expf(t));
  }
}

// ---------------------------------------------------------------------------
extern "C" void kernel_launch(void* const* d_in, const int* in_sizes, int n_in,
                              void* d_out, int out_size, void* d_ws, size_t ws_size,
                              hipStream_t stream) {
  const float* node_features = (const float*)d_in[0];
  const float* edge_attr     = (const float*)d_in[1];
  const int*   edge_index    = (const int*)d_in[2];
  const float* W_node = (const float*)d_in[3];
  const float* b_node = (const float*)d_in[4];
  const float* W_edge = (const float*)d_in[5];
  const float* b_edge = (const float*)d_in[6];
  const float* W_ih   = (const float*)d_in[7];
  const float* b_ih   = (const float*)d_in[8];
  const float* W_hh   = (const float*)d_in[9];
  const float* b_hh   = (const float*)d_in[10];
  const float* W_dec1 = (const float*)d_in[11];
  const float* b_dec1 = (const float*)d_in[12];
  const float* W_dec2 = (const float*)d_in[13];
  const float* b_dec2 = (const float*)d_in[14];
  float* logits = (float*)d_out;

  const size_t NH = (size_t)N_NODES * HID;
  float* x    = (float*)d_ws;
  float* xn   = x + NH;
  float* agg  = xn + NH;

  encode_nodes_kernel<<<(N_NODES * HID + 255) / 256, 256, 0, stream>>>(
      node_features, W_node, b_node, x);

  const long EF = (long)N_EDGES * HID;
  for (int depth = 0; depth < DEPTH; ++depth) {
    zero_kernel<<<(N_NODES * HID + 255) / 256, 256, 0, stream>>>(agg, N_NODES * HID);
    scatter_msg_kernel<<<(int)((EF + 255) / 256), 256, 0, stream>>>(
        x, edge_attr, edge_index, W_edge, b_edge, agg);
    gru_kernel<<<N_NODES / 16, 32, 0, stream>>>(
        agg, x, W_ih, b_ih, W_hh, b_hh, xn);
    float* tmp = x; x = xn; xn = tmp;
  }

  decoder_kernel<<<N_EDGES / 16, 32, 0, stream>>>(
      x, edge_attr, edge_index, W_edge, b_edge,
      W_dec1, b_dec1, W_dec2, b_dec2, logits);
}